// Pooling_89223650607320
// MI455X (gfx1250) — compile-verified
//
#include <hip/hip_runtime.h>
#include <math.h>

#define BATCH 64
#define SEQ   2048
#define DIM   512
#define HID   128
#define TOPK  2

// ---- kernel 1 tiling ----
#define ROWS  64            // rows of x(BL, D) per block
#define RS    (DIM + 4)     // padded LDS row stride (floats) -> avoids 64-bank conflicts
#define NWAVE 8             // 256 threads / 32

// ---- kernel 3 chunking ----
#define CH    8
#define LCH   (SEQ / CH)

typedef float v2f __attribute__((ext_vector_type(2)));
typedef float v8f __attribute__((ext_vector_type(8)));

// =====================================================================
// Kernel 1: scores via WMMA FP32 GEMM.  s_att = relu(x@W1+b1)@W2+b2,
//           s_latt = x@w_latt+b_latt.  One pass over x.
// =====================================================================
__global__ __launch_bounds__(256) void ffn_score_kernel(
    const float* __restrict__ x,  const float* __restrict__ W1,
    const float* __restrict__ b1, const float* __restrict__ W2,
    const float* __restrict__ b2, const float* __restrict__ wl,
    const float* __restrict__ blv,
    float* __restrict__ s_att, float* __restrict__ s_latt)
{
    __shared__ float lx[ROWS * RS];          // 64 x 516 f32 = 132 KB
    __shared__ float part[ROWS][NWAVE];      // per-wave s_att partials

    const int tid  = threadIdx.x;
    const int wave = tid >> 5;
    const int lane = tid & 31;
    const int half = lane >> 4;              // 0: lanes 0-15, 1: lanes 16-31
    const int l16  = lane & 15;
    const size_t row0 = (size_t)blockIdx.x * ROWS;

    // -------- cooperative load: contiguous 64x512 slab -> padded LDS --------
    {
        const float4* src = (const float4*)(x + row0 * DIM);
        for (int i = tid; i < ROWS * (DIM / 4); i += 256) {
            const int r = i >> 7;            // DIM/4 == 128
            const int c = i & 127;
            ((float4*)(lx + r * RS))[c] = src[i];
        }
    }
    __syncthreads();

    // -------- WMMA f32 16x16x4: wave owns 16-col tile, 4 row slabs --------
    const int c0 = wave * 16;                // column base in H
    const float* Wcol = W1 + c0 + l16;       // W1 is (D, H) row-major

    v8f acc0 = {}, acc1 = {}, acc2 = {}, acc3 = {};
    for (int kk = 0; kk < DIM; kk += 4) {
        const int kb = kk + 2 * half;        // K pair selected by lane half
        v2f bf;
        bf.x = Wcol[(size_t)(kb + 0) * HID]; // B frag: vgpr0=K, vgpr1=K+1
        bf.y = Wcol[(size_t)(kb + 1) * HID];
        const float* abase = &lx[l16 * RS + kb];
        v2f a0 = *(const v2f*)(abase + 0 * 16 * RS);
        v2f a1 = *(const v2f*)(abase + 1 * 16 * RS);
        v2f a2 = *(const v2f*)(abase + 2 * 16 * RS);
        v2f a3 = *(const v2f*)(abase + 3 * 16 * RS);
        acc0 = __builtin_amdgcn_wmma_f32_16x16x4_f32(false, a0, false, bf, (short)0, acc0, false, false);
        acc1 = __builtin_amdgcn_wmma_f32_16x16x4_f32(false, a1, false, bf, (short)0, acc1, false, false);
        acc2 = __builtin_amdgcn_wmma_f32_16x16x4_f32(false, a2, false, bf, (short)0, acc2, false, false);
        acc3 = __builtin_amdgcn_wmma_f32_16x16x4_f32(false, a3, false, bf, (short)0, acc3, false, false);
    }

    // -------- bias + relu, fold with W2 (per-row sum over this wave's 16 cols)
    const float w2c = W2[c0 + l16];
    const float bc1 = b1[c0 + l16];
    {
        v8f accs[4] = {acc0, acc1, acc2, acc3};
#pragma unroll
        for (int t = 0; t < 4; ++t) {
#pragma unroll
            for (int v = 0; v < 8; ++v) {
                float h = accs[t][v] + bc1;
                h = fmaxf(h, 0.0f);
                float p = h * w2c;
                p += __shfl_xor(p, 1);
                p += __shfl_xor(p, 2);
                p += __shfl_xor(p, 4);
                p += __shfl_xor(p, 8);       // sum over 16 lanes in this half
                if (l16 == 0) part[t * 16 + v + 8 * half][wave] = p;
            }
        }
    }
    __syncthreads();

    if (tid < ROWS) {
        float s = b2[0];
#pragma unroll
        for (int w = 0; w < NWAVE; ++w) s += part[tid][w];
        s_att[row0 + tid] = s;
    }

    // -------- s_latt = x . w_latt, straight from the LDS slab --------
    for (int r = wave; r < ROWS; r += NWAVE) {
        float s = 0.0f;
        const float* rowp = &lx[r * RS];
        for (int k = lane; k < DIM; k += 32) s += rowp[k] * wl[k];
        s += __shfl_xor(s, 1);
        s += __shfl_xor(s, 2);
        s += __shfl_xor(s, 4);
        s += __shfl_xor(s, 8);
        s += __shfl_xor(s, 16);
        if (lane == 0) s_latt[row0 + r] = s + blv[0];
    }
}

// =====================================================================
// Kernel 2: masked softmax over L for both score arrays -> alpha arrays
// =====================================================================
__global__ __launch_bounds__(256) void softmax_kernel(
    const float* __restrict__ s_att, const float* __restrict__ s_latt,
    const int* __restrict__ lengths,
    float* __restrict__ a_att, float* __restrict__ a_latt)
{
    __shared__ float red[256];
    const int b   = blockIdx.x;
    const int tid = threadIdx.x;
    const int len = lengths[b];

    for (int j = 0; j < 2; ++j) {
        const float* s = (j == 0 ? s_att : s_latt) + (size_t)b * SEQ;
        float*       a = (j == 0 ? a_att : a_latt) + (size_t)b * SEQ;

        float m = -INFINITY;
        for (int l = tid; l < len; l += 256) m = fmaxf(m, s[l]);
        red[tid] = m; __syncthreads();
        for (int o = 128; o > 0; o >>= 1) {
            if (tid < o) red[tid] = fmaxf(red[tid], red[tid + o]);
            __syncthreads();
        }
        m = red[0]; __syncthreads();

        float sum = 0.0f;
        for (int l = tid; l < len; l += 256) sum += __expf(s[l] - m);
        red[tid] = sum; __syncthreads();
        for (int o = 128; o > 0; o >>= 1) {
            if (tid < o) red[tid] += red[tid + o];
            __syncthreads();
        }
        const float inv = 1.0f / red[0]; __syncthreads();

        for (int l = tid; l < SEQ; l += 256)
            a[l] = (l < len) ? __expf(s[l] - m) * inv : 0.0f;
        __syncthreads();
    }
}

// =====================================================================
// Kernel 3: second pass over x.  One thread per d; chunked over L.
// Produces per-chunk partial weighted sums + top-2 running maxima.
// =====================================================================
__global__ __launch_bounds__(512) void stream_kernel(
    const float* __restrict__ x, const float* __restrict__ a_att,
    const float* __restrict__ a_latt, const int* __restrict__ lengths,
    float* __restrict__ pAtt, float* __restrict__ pLatt,
    float* __restrict__ pM1, float* __restrict__ pM2)
{
    const int bc = blockIdx.x;
    const int b  = bc / CH;
    const int ch = bc % CH;
    const int d  = threadIdx.x;
    const int len = lengths[b];
    const int l0  = ch * LCH;

    const float* xp = x + ((size_t)b * SEQ + l0) * DIM + d;
    const float* aa = a_att  + (size_t)b * SEQ + l0;
    const float* al = a_latt + (size_t)b * SEQ + l0;
    const int lim = min(LCH, len - l0);      // valid count in this chunk (<=0 ok)

    float sA = 0.0f, sL = 0.0f, m1 = -INFINITY, m2 = -INFINITY;
    for (int i = 0; i < LCH; ++i) {
        const float xv = xp[(size_t)i * DIM];
        sA = fmaf(aa[i], xv, sA);            // alpha==0 past len -> no masking
        sL = fmaf(al[i], xv, sL);
        if (i < lim) {
            if (xv > m1)      { m2 = m1; m1 = xv; }
            else if (xv > m2) { m2 = xv; }
        }
    }
    const size_t o = (size_t)bc * DIM + d;
    pAtt[o] = sA; pLatt[o] = sL; pM1[o] = m1; pM2[o] = m2;
}

// =====================================================================
// Kernel 4: merge chunk partials, fetch last row, write (B, 6D) concat:
// [p_max | p_last | topk0 | topk1 | p_att | p_latt]
// =====================================================================
__global__ __launch_bounds__(512) void finalize_kernel(
    const float* __restrict__ x, const int* __restrict__ lengths,
    const float* __restrict__ pAtt, const float* __restrict__ pLatt,
    const float* __restrict__ pM1, const float* __restrict__ pM2,
    float* __restrict__ out)
{
    const int b = blockIdx.x;
    const int d = threadIdx.x;

    float sA = 0.0f, sL = 0.0f, m1 = -INFINITY, m2 = -INFINITY;
#pragma unroll
    for (int ch = 0; ch < CH; ++ch) {
        const size_t o = ((size_t)b * CH + ch) * DIM + d;
        sA += pAtt[o];
        sL += pLatt[o];
        const float c1 = pM1[o], c2 = pM2[o];
        if (c1 > m1) { m2 = fmaxf(m1, c2); m1 = c1; }
        else         { m2 = fmaxf(m2, c1); }
    }
    const int len = lengths[b];
    const float last = x[((size_t)b * SEQ + (len - 1)) * DIM + d];

    float* ob = out + (size_t)b * (6 * DIM);
    ob[0 * DIM + d] = m1;    // p_max
    ob[1 * DIM + d] = last;  // p_last
    ob[2 * DIM + d] = m1;    // topk k=0
    ob[3 * DIM + d] = m2;    // topk k=1
    ob[4 * DIM + d] = sA;    // p_att
    ob[5 * DIM + d] = sL;    // p_latt
}

// =====================================================================
extern "C" void kernel_launch(void* const* d_in, const int* in_sizes, int n_in,
                              void* d_out, int out_size, void* d_ws, size_t ws_size,
                              hipStream_t stream)
{
    // setup_inputs() dict order: x, W1, b1, W2, b2, w_latt, b_latt, lengths
    const float* x   = (const float*)d_in[0];
    const float* W1  = (const float*)d_in[1];
    const float* b1  = (const float*)d_in[2];
    const float* W2  = (const float*)d_in[3];
    const float* b2  = (const float*)d_in[4];
    const float* wl  = (const float*)d_in[5];
    const float* blv = (const float*)d_in[6];
    const int*   len = (const int*)d_in[7];
    float* out = (float*)d_out;

    // workspace carve (floats): 4x (B*L) scores/alphas + 4x (B*CH*D) partials
    const size_t BL  = (size_t)BATCH * SEQ;        // 131072
    const size_t BCD = (size_t)BATCH * CH * DIM;   // 262144
    float* w = (float*)d_ws;
    float* s_att  = w;              // BL
    float* s_latt = w + BL;         // BL
    float* a_att  = w + 2 * BL;     // BL
    float* a_latt = w + 3 * BL;     // BL
    float* pAtt   = w + 4 * BL;             // BCD
    float* pLatt  = w + 4 * BL + 1 * BCD;   // BCD
    float* pM1    = w + 4 * BL + 2 * BCD;   // BCD
    float* pM2    = w + 4 * BL + 3 * BCD;   // BCD

    const int nblk1 = (int)(BL / ROWS);     // 2048
    ffn_score_kernel<<<nblk1, 256, 0, stream>>>(x, W1, b1, W2, b2, wl, blv,
                                                s_att, s_latt);
    softmax_kernel<<<BATCH, 256, 0, stream>>>(s_att, s_latt, len, a_att, a_latt);
    stream_kernel<<<BATCH * CH, 512, 0, stream>>>(x, a_att, a_latt, len,
                                                  pAtt, pLatt, pM1, pM2);
    finalize_kernel<<<BATCH, 512, 0, stream>>>(x, len, pAtt, pLatt, pM1, pM2, out);
}